// VoxelBackBone8x_62697932587673
// MI455X (gfx1250) — compile-verified
//
#include <hip/hip_runtime.h>

// ---------------------------------------------------------------------------
// CDNA5 (gfx1250) implementation of VoxelBackBone8x (dense path).
// Convs: implicit GEMM on V_WMMA_F32_16X16X4_F32 (fp32 in/out, K=4 Cin tile).
// Each wave: 64 spatial positions x 16 out-channels, 4 named accumulators.
// Weights for the block's 16 out-channels staged in LDS (channel-pair
// interleaved: one ds_load_b64 per B fragment).
// A loads are unconditional (offset clamped, value masked 0/1), use a single
// shared unsigned voffset against two SGPR base pointers (SADDR+scale_offset),
// and are issued in row-granularity load/compute phases so the scheduler
// emits long VMEM clauses with partial s_wait_loadcnt.
// Workload is HBM-bound (~45 GFLOP vs ~0.5 GB @ 23.3 TB/s) -> fp32 WMMA keeps
// numerics aligned with the fp32 JAX reference at no bandwidth cost.
// ---------------------------------------------------------------------------

typedef float v2f __attribute__((ext_vector_type(2)));
typedef float v8f __attribute__((ext_vector_type(8)));

#define EPSV 1e-3f

// ---------------------------------------------------------------------------
// Scatter voxels into dense NCDHW grid + occupancy mask.
// ---------------------------------------------------------------------------
__global__ void scatter_kernel(const float* __restrict__ vf,
                               const int* __restrict__ coords,
                               float* __restrict__ dense,
                               float* __restrict__ mask,
                               int nvox, int Cin, int D, int H, int W) {
  int i = blockIdx.x * blockDim.x + threadIdx.x;
  if (i >= nvox) return;
  int z = coords[i * 4 + 1];
  int y = coords[i * 4 + 2];
  int x = coords[i * 4 + 3];
  int S = D * H * W;
  int pos = (z * H + y) * W + x;
  for (int c = 0; c < Cin; ++c) dense[c * S + pos] = vf[i * Cin + c];
  mask[pos] = 1.0f;
}

// ---------------------------------------------------------------------------
// Mask downsample: conv with ones kernel, >0 -> 1.
// ---------------------------------------------------------------------------
__global__ void mask_down_kernel(const float* __restrict__ mi, float* __restrict__ mo,
                                 int D, int H, int W, int D2, int H2, int W2,
                                 int KD, int KH, int KW, int sd, int sh, int sw,
                                 int pd, int ph, int pw) {
  int pos = blockIdx.x * blockDim.x + threadIdx.x;
  int S2 = D2 * H2 * W2;
  if (pos >= S2) return;
  int z = pos / (H2 * W2);
  int r = pos - z * (H2 * W2);
  int y = r / W2;
  int x = r - y * W2;
  float v = 0.f;
  for (int kd = 0; kd < KD; ++kd) {
    int iz = z * sd - pd + kd;
    if ((unsigned)iz >= (unsigned)D) continue;
    for (int kh = 0; kh < KH; ++kh) {
      int iy = y * sh - ph + kh;
      if ((unsigned)iy >= (unsigned)H) continue;
      for (int kw = 0; kw < KW; ++kw) {
        int ix = x * sw - pw + kw;
        if ((unsigned)ix >= (unsigned)W) continue;
        v += mi[(iz * H + iy) * W + ix];
      }
    }
  }
  mo[pos] = (v > 0.f) ? 1.f : 0.f;
}

// ---------------------------------------------------------------------------
// Mask population count (block partial -> one atomic per block).
// ---------------------------------------------------------------------------
__global__ void mask_count_kernel(const float* __restrict__ mask, int S,
                                  float* __restrict__ cnt) {
  __shared__ float sm[256];
  int tid = threadIdx.x;
  float s = 0.f;
  for (int i = blockIdx.x * 256 + tid; i < S; i += gridDim.x * 256) s += mask[i];
  sm[tid] = s;
  __syncthreads();
  for (int o = 128; o > 0; o >>= 1) {
    if (tid < o) sm[tid] += sm[tid + o];
    __syncthreads();
  }
  if (tid == 0) atomicAdd(cnt, sm[0]);
}

// ---------------------------------------------------------------------------
// Implicit-GEMM 3D conv on V_WMMA_F32_16X16X4_F32.
// Fragment layouts (ISA 7.12.2):
//   A (16x4 f32): lane L: row M = L&15; VGPR v holds K = (L>>4)*2 + v.
//   B (4x16 f32): lane L: col N = L&15; VGPR v holds K = (L>>4)*2 + v.
//   C/D: lane L, VGPR v = (M = v + 8*(L>>4), N = L&15) -> 8 contiguous floats.
// LDS weight layout: pairs of input channels interleaved:
//   wsh[(((k>>1)*KTAP + tap)*16 + n)*2 + (k&1)]  -> B fragment = one b64 load.
// Inner loop is phased per (kd,kh) row: issue all KW*4 A-pairs + KW B-pairs
// first (pure load clause), then 4*KW independent WMMAs.
// ---------------------------------------------------------------------------
template <int KD, int KH, int KW, int CIN>
__global__ __launch_bounds__(128) void conv3d_wmma(
    const float* __restrict__ in, const float* __restrict__ w,
    float* __restrict__ out, int Cout,
    int D, int H, int W, int D2, int H2, int W2,
    int sd, int sh, int sw, int pd, int ph, int pw) {
  constexpr int KTAP = KD * KH * KW;
  constexpr int WTOT = CIN * KTAP * 16;
  __shared__ float wsh[WTOT];

  const int tid = threadIdx.x;
  const int lane = tid & 31;
  const int wid = tid >> 5;
  const int nbase0 = blockIdx.y * 16;
  const int S2 = D2 * H2 * W2;
  const int S1 = D * H * W;

  // Cooperative weight staging (weights are OIDHW), channel-pair interleaved.
  for (int idx = tid; idx < WTOT; idx += 128) {
    int n = idx / (CIN * KTAP);
    int rem = idx - n * (CIN * KTAP);
    int k = rem / KTAP;
    int tap = rem - k * KTAP;
    wsh[(((k >> 1) * KTAP + tap) * 16 + n) * 2 + (k & 1)] =
        w[(long)((nbase0 + n) * CIN + k) * KTAP + tap];
  }
  __syncthreads();

  const int mblk = (blockIdx.x * 4 + wid) * 64;
  if (mblk >= S2) return;  // wave-uniform (after the only barrier)

  const int m = lane & 15;
  const int hi = lane >> 4;  // half-wave id: supplies K rows 2*hi, 2*hi+1
  const int n = lane & 15;

  // Two SGPR base pointers: channel c and channel c+1 planes.
  const float* __restrict__ in0 = in;
  const float* __restrict__ in1 = in + S1;

  // Per-subtile output coordinates for this lane's A row.
  int izA[4], iyA[4], ixA[4], ibase[4], mvA[4];
#pragma unroll
  for (int t = 0; t < 4; ++t) {
    int pos = mblk + t * 16 + m;
    int valid = pos < S2;
    int pc = valid ? pos : 0;
    int pz = pc / (H2 * W2);
    int r = pc - pz * (H2 * W2);
    int py = r / W2;
    int px = r - py * W2;
    izA[t] = pz * sd - pd;
    iyA[t] = py * sh - ph;
    ixA[t] = px * sw - pw;
    ibase[t] = (izA[t] * H + iyA[t]) * W + ixA[t];
    mvA[t] = valid;
  }

  v8f acc0 = {}, acc1 = {}, acc2 = {}, acc3 = {};
  const v2f* __restrict__ wp = (const v2f*)wsh;

  for (int k0 = 0; k0 < CIN; k0 += 4) {
    const unsigned abase = (unsigned)((k0 + 2 * hi) * S1);  // per-lane channel base
    const int wrow = ((k0 >> 1) + hi) * KTAP * 16 + n;      // v2f index base
#pragma unroll
    for (int kd = 0; kd < KD; ++kd) {
#pragma unroll
      for (int kh = 0; kh < KH; ++kh) {
        // ---- phase 1: pure load clause (no ALU consumers in between) ----
        v2f bb[KW];                  // B fragments for the row
        float r0[KW][4], r1[KW][4];  // raw A values (unmasked)
        float mm[KW][4];             // 0/1 masks
#pragma unroll
        for (int kw = 0; kw < KW; ++kw) {
          const int tap = (kd * KH + kh) * KW + kw;
          bb[kw] = wp[wrow + tap * 16];  // ds_load_b64
#pragma unroll
          for (int t = 0; t < 4; ++t) {
            int iz = izA[t] + kd, iy = iyA[t] + kh, ix = ixA[t] + kw;
            int inb = mvA[t] & ((unsigned)iz < (unsigned)D) &
                      ((unsigned)iy < (unsigned)H) & ((unsigned)ix < (unsigned)W);
            int ib = ibase[t] + (kd * H + kh) * W + kw;
            // Clamp: in-bounds => ib in [0,S1); OOB => 0. Value masked later.
            unsigned offv = abase + (unsigned)(inb ? ib : 0);
            mm[kw][t] = inb ? 1.0f : 0.0f;
            r0[kw][t] = in0[offv];  // global_load_b32 v,v,s[..] scale_offset
            r1[kw][t] = in1[offv];  // same voffset, second SGPR base
          }
        }
        // ---- phase 2: 4*KW independent WMMAs (partial waits suffice) ----
#pragma unroll
        for (int kw = 0; kw < KW; ++kw) {
          v2f a0, a1, a2, a3;
          a0.x = r0[kw][0] * mm[kw][0]; a0.y = r1[kw][0] * mm[kw][0];
          a1.x = r0[kw][1] * mm[kw][1]; a1.y = r1[kw][1] * mm[kw][1];
          a2.x = r0[kw][2] * mm[kw][2]; a2.y = r1[kw][2] * mm[kw][2];
          a3.x = r0[kw][3] * mm[kw][3]; a3.y = r1[kw][3] * mm[kw][3];
          acc0 = __builtin_amdgcn_wmma_f32_16x16x4_f32(
              false, a0, false, bb[kw], (short)0, acc0, false, false);
          acc1 = __builtin_amdgcn_wmma_f32_16x16x4_f32(
              false, a1, false, bb[kw], (short)0, acc1, false, false);
          acc2 = __builtin_amdgcn_wmma_f32_16x16x4_f32(
              false, a2, false, bb[kw], (short)0, acc2, false, false);
          acc3 = __builtin_amdgcn_wmma_f32_16x16x4_f32(
              false, a3, false, bb[kw], (short)0, acc3, false, false);
        }
      }
    }
  }

  // Store: lane holds rows M = v + 8*hi at fixed out-channel.
  const int oc = nbase0 + n;
  auto st = [&](int t, const v8f& acc) {
    int pbase = mblk + t * 16 + 8 * hi;
    long obase = (long)oc * S2 + pbase;
    if (mblk + t * 16 + 15 < S2) {
      *(v8f*)(out + obase) = acc;  // 32B-aligned, 2x global_store_b128
    } else {
#pragma unroll
      for (int v = 0; v < 8; ++v) {
        if (pbase + v < S2) out[obase + v] = acc[v];
      }
    }
  };
  st(0, acc0);
  st(1, acc1);
  st(2, acc2);
  st(3, acc3);
}

// ---------------------------------------------------------------------------
// Masked BN: per-channel sum / sumsq over masked positions.
// ---------------------------------------------------------------------------
__global__ void bn_reduce_kernel(const float* __restrict__ y,
                                 const float* __restrict__ mask, int S,
                                 float* __restrict__ sums,
                                 float* __restrict__ sumsq) {
  __shared__ float s1[256];
  __shared__ float s2[256];
  int c = blockIdx.y;
  const float* yc = y + (long)c * S;
  int tid = threadIdx.x;
  float a = 0.f, b = 0.f;
  for (int i = blockIdx.x * 256 + tid; i < S; i += gridDim.x * 256) {
    float mv = mask[i];
    float v = yc[i] * mv;  // mask is {0,1}: v*yc == y^2*m
    a += v;
    b += v * yc[i];
  }
  s1[tid] = a;
  s2[tid] = b;
  __syncthreads();
  for (int o = 128; o > 0; o >>= 1) {
    if (tid < o) {
      s1[tid] += s1[tid + o];
      s2[tid] += s2[tid + o];
    }
    __syncthreads();
  }
  if (tid == 0) {
    atomicAdd(&sums[c], s1[0]);
    atomicAdd(&sumsq[c], s2[0]);
  }
}

__global__ void bn_finalize_kernel(const float* __restrict__ sums,
                                   const float* __restrict__ sumsq,
                                   const float* __restrict__ cnt,
                                   const float* __restrict__ gamma,
                                   const float* __restrict__ beta,
                                   float* __restrict__ scl,
                                   float* __restrict__ shf, int C) {
  int c = blockIdx.x * blockDim.x + threadIdx.x;
  if (c >= C) return;
  float nvox = fmaxf(cnt[0], 1.0f);
  float mean = sums[c] / nvox;
  float var = sumsq[c] / nvox - mean * mean;
  float inv = rsqrtf(var + EPSV);
  float a = inv * gamma[c];
  scl[c] = a;
  shf[c] = beta[c] - mean * a;
}

__global__ void bn_apply_kernel(float* __restrict__ y,
                                const float* __restrict__ mask,
                                const float* __restrict__ scl,
                                const float* __restrict__ shf, int C, int S) {
  long i = (long)blockIdx.x * blockDim.x + threadIdx.x;
  long tot = (long)C * S;
  if (i >= tot) return;
  int c = (int)(i / S);
  int p = (int)(i - (long)c * S);
  float v = y[i] * scl[c] + shf[c];
  v = fmaxf(v, 0.0f) * mask[p];
  y[i] = v;
}

// ---------------------------------------------------------------------------
// Host orchestration.
// ---------------------------------------------------------------------------
extern "C" void kernel_launch(void* const* d_in, const int* in_sizes, int n_in,
                              void* d_out, int out_size, void* d_ws, size_t ws_size,
                              hipStream_t stream) {
  (void)in_sizes; (void)n_in; (void)out_size; (void)ws_size;

  const float* vf = (const float*)d_in[0];
  const int* coords = (const int*)d_in[1];
  auto Wp = [&](int i) { return (const float*)d_in[2 + 3 * i]; };
  auto Gp = [&](int i) { return (const float*)d_in[3 + 3 * i]; };
  auto Bp = [&](int i) { return (const float*)d_in[4 + 3 * i]; };

  const int NVOX = 60000, CIN = 4;
  const int D1 = 41, H1 = 160, W1 = 160;
  const int D2 = 21, H2 = 80, W2 = 80;
  const int D3 = 11, H3 = 40, W3 = 40;
  const int D4 = 5, H4 = 20, W4 = 20;
  const int D5 = 2, H5 = 20, W5 = 20;
  const int S1 = D1 * H1 * W1;  // 1,049,600
  const int S2 = D2 * H2 * W2;  // 134,400
  const int S3 = D3 * H3 * W3;  // 17,600
  const int S4 = D4 * H4 * W4;  // 2,000
  const int S5 = D5 * H5 * W5;  // 800

  // Workspace carving (256B aligned).
  char* ws = (char*)d_ws;
  size_t off = 0;
  auto alloc = [&](size_t nfloats) {
    float* p = (float*)(ws + off);
    off += ((nfloats * sizeof(float) + 255) / 256) * 256;
    return p;
  };
  float* dense = alloc((size_t)CIN * S1);
  float* bufA = alloc((size_t)16 * S1);
  float* bufB = alloc((size_t)16 * S1);
  float* m1 = alloc(S1);
  float* m2 = alloc(S2);
  float* m3 = alloc(S3);
  float* m4 = alloc(S4);
  float* m5 = alloc(S5);
  float* sums = alloc(128);
  float* sumsq = alloc(128);
  float* scl = alloc(128);
  float* shf = alloc(128);
  float* cnts = alloc(8);

  // Init.
  hipMemsetAsync(dense, 0, (size_t)CIN * S1 * sizeof(float), stream);
  hipMemsetAsync(m1, 0, (size_t)S1 * sizeof(float), stream);
  hipMemsetAsync(cnts, 0, 8 * sizeof(float), stream);

  scatter_kernel<<<(NVOX + 255) / 256, 256, 0, stream>>>(vf, coords, dense, m1,
                                                         NVOX, CIN, D1, H1, W1);

  // Mask pyramid.
  mask_down_kernel<<<(S2 + 255) / 256, 256, 0, stream>>>(
      m1, m2, D1, H1, W1, D2, H2, W2, 3, 3, 3, 2, 2, 2, 1, 1, 1);
  mask_down_kernel<<<(S3 + 255) / 256, 256, 0, stream>>>(
      m2, m3, D2, H2, W2, D3, H3, W3, 3, 3, 3, 2, 2, 2, 1, 1, 1);
  mask_down_kernel<<<(S4 + 255) / 256, 256, 0, stream>>>(
      m3, m4, D3, H3, W3, D4, H4, W4, 3, 3, 3, 2, 2, 2, 0, 1, 1);
  mask_down_kernel<<<(S5 + 255) / 256, 256, 0, stream>>>(
      m4, m5, D4, H4, W4, D5, H5, W5, 3, 1, 1, 2, 1, 1, 0, 0, 0);

  // Mask counts.
  auto count_mask = [&](const float* m, int S, float* c) {
    int blk = (S + 256 * 8 - 1) / (256 * 8);
    if (blk < 1) blk = 1;
    if (blk > 256) blk = 256;
    mask_count_kernel<<<blk, 256, 0, stream>>>(m, S, c);
  };
  count_mask(m1, S1, cnts + 0);
  count_mask(m2, S2, cnts + 1);
  count_mask(m3, S3, cnts + 2);
  count_mask(m4, S4, cnts + 3);
  count_mask(m5, S5, cnts + 4);

  // Conv dispatch over template instantiations.
  auto launch_conv = [&](const float* in, const float* w, float* out, int Cin,
                         int Cout, int Di, int Hi, int Wi, int Do, int Ho,
                         int Wo, int KD, int sd, int shh, int sw, int pd,
                         int ph, int pw) {
    int So = Do * Ho * Wo;
    dim3 grid((So + 255) / 256, Cout / 16);  // 4 waves x 64 positions / block
    if (KD == 3 && sw == 1 && ph == 0 && pw == 0) {
      // conv_out: kernel (3,1,1)
      conv3d_wmma<3, 1, 1, 64><<<grid, 128, 0, stream>>>(
          in, w, out, Cout, Di, Hi, Wi, Do, Ho, Wo, sd, shh, sw, pd, ph, pw);
    } else if (Cin == 4) {
      conv3d_wmma<3, 3, 3, 4><<<grid, 128, 0, stream>>>(
          in, w, out, Cout, Di, Hi, Wi, Do, Ho, Wo, sd, shh, sw, pd, ph, pw);
    } else if (Cin == 16) {
      conv3d_wmma<3, 3, 3, 16><<<grid, 128, 0, stream>>>(
          in, w, out, Cout, Di, Hi, Wi, Do, Ho, Wo, sd, shh, sw, pd, ph, pw);
    } else if (Cin == 32) {
      conv3d_wmma<3, 3, 3, 32><<<grid, 128, 0, stream>>>(
          in, w, out, Cout, Di, Hi, Wi, Do, Ho, Wo, sd, shh, sw, pd, ph, pw);
    } else {
      conv3d_wmma<3, 3, 3, 64><<<grid, 128, 0, stream>>>(
          in, w, out, Cout, Di, Hi, Wi, Do, Ho, Wo, sd, shh, sw, pd, ph, pw);
    }
  };

  // One conv + masked-BN-ReLU layer.
  auto run_layer = [&](const float* in, float* out, int li, int Cin, int Cout,
                       int Di, int Hi, int Wi, int Do, int Ho, int Wo, int KD,
                       int sd, int shh, int sw, int pd, int ph, int pw,
                       const float* mask, const float* cntp) {
    int So = Do * Ho * Wo;
    launch_conv(in, Wp(li), out, Cin, Cout, Di, Hi, Wi, Do, Ho, Wo, KD, sd,
                shh, sw, pd, ph, pw);
    hipMemsetAsync(sums, 0, Cout * sizeof(float), stream);
    hipMemsetAsync(sumsq, 0, Cout * sizeof(float), stream);
    int bpc = (So + 256 * 8 - 1) / (256 * 8);
    if (bpc < 1) bpc = 1;
    if (bpc > 256) bpc = 256;
    bn_reduce_kernel<<<dim3(bpc, Cout), 256, 0, stream>>>(out, mask, So, sums,
                                                          sumsq);
    bn_finalize_kernel<<<(Cout + 63) / 64, 64, 0, stream>>>(
        sums, sumsq, cntp, Gp(li), Bp(li), scl, shf, Cout);
    long tot = (long)Cout * So;
    bn_apply_kernel<<<(unsigned)((tot + 255) / 256), 256, 0, stream>>>(
        out, mask, scl, shf, Cout, So);
  };

  // Layer schedule (spec order: in,1,2d,2a,2b,3d,3a,3b,4d,4a,4b,out).
  run_layer(dense, bufA, 0, 4, 16, D1, H1, W1, D1, H1, W1, 3, 1, 1, 1, 1, 1, 1, m1, cnts + 0);
  run_layer(bufA, bufB, 1, 16, 16, D1, H1, W1, D1, H1, W1, 3, 1, 1, 1, 1, 1, 1, m1, cnts + 0);
  run_layer(bufB, bufA, 2, 16, 32, D1, H1, W1, D2, H2, W2, 3, 2, 2, 2, 1, 1, 1, m2, cnts + 1);
  run_layer(bufA, bufB, 3, 32, 32, D2, H2, W2, D2, H2, W2, 3, 1, 1, 1, 1, 1, 1, m2, cnts + 1);
  run_layer(bufB, bufA, 4, 32, 32, D2, H2, W2, D2, H2, W2, 3, 1, 1, 1, 1, 1, 1, m2, cnts + 1);
  run_layer(bufA, bufB, 5, 32, 64, D2, H2, W2, D3, H3, W3, 3, 2, 2, 2, 1, 1, 1, m3, cnts + 2);
  run_layer(bufB, bufA, 6, 64, 64, D3, H3, W3, D3, H3, W3, 3, 1, 1, 1, 1, 1, 1, m3, cnts + 2);
  run_layer(bufA, bufB, 7, 64, 64, D3, H3, W3, D3, H3, W3, 3, 1, 1, 1, 1, 1, 1, m3, cnts + 2);
  run_layer(bufB, bufA, 8, 64, 64, D3, H3, W3, D4, H4, W4, 3, 2, 2, 2, 0, 1, 1, m4, cnts + 3);
  run_layer(bufA, bufB, 9, 64, 64, D4, H4, W4, D4, H4, W4, 3, 1, 1, 1, 1, 1, 1, m4, cnts + 3);
  run_layer(bufB, bufA, 10, 64, 64, D4, H4, W4, D4, H4, W4, 3, 1, 1, 1, 1, 1, 1, m4, cnts + 3);
  run_layer(bufA, (float*)d_out, 11, 64, 128, D4, H4, W4, D5, H5, W5, 3, 2, 1, 1, 0, 0, 0, m5, cnts + 4);
}